// TCCLModel_79774722556565
// MI455X (gfx1250) — compile-verified
//
#include <hip/hip_runtime.h>
#include <hip/hip_bf16.h>
#include <math.h>
#include <cstddef>

typedef __attribute__((ext_vector_type(2))) float v2f;
typedef __attribute__((ext_vector_type(8))) float v8f;

#define EPS_BN 1e-5f
#define EPS_NORM 1e-12f
#define INV_T 10.0f   // 1/TEMPERATURE

// ---------------- workspace layout (float offsets) ----------------
static constexpr size_t OFF_A   = 0;          // big scratch region: y1/y2/y3/t1/t2
static constexpr size_t OFF_B   = 33554432;   // h1 / h2 region (8,388,608 floats)
static constexpr size_t OFF_H3  = 41943040;   // 2 x 2,097,152
static constexpr size_t OFF_K   = 46137344;   // 2 x 81,920 (B,64,5)
static constexpr size_t OFF_U   = 46301184;   // 2 x 65,536 (256 x 256 pack from k)
static constexpr size_t OFF_V   = 46432256;   // 2 x 65,536 (256 x 256 pack from h)
static constexpr size_t OFF_AN  = 46563328;   // 2 x 81,920 (l2-normed flat k, 256x320)
static constexpr size_t OFF_SIM = 46727168;   // 3 x 65,536
static constexpr size_t OFF_RL  = 46923776;   // 3 x 256 row losses
static constexpr size_t OFF_ST  = 46924544;   // 128 stats (sum, sumsq)
static constexpr size_t OFF_WP  = 46924672;   // packed (K->Kp zero-padded) conv weights
// packed weight sub-offsets
static constexpr size_t OFF_W2P  = OFF_WP;                 // 32*16*20 = 10240
static constexpr size_t OFF_W3P  = OFF_W2P + 10240;        // 64*32*20 = 40960
static constexpr size_t OFF_TW1P = OFF_W3P + 40960;        // 64*64*8  = 32768
static constexpr size_t OFF_TW2P = OFF_TW1P + 32768;       // 64*64*4  = 16384
// scratch-A sub-offsets for template head
static constexpr size_t OFF_T1  = OFF_A;
static constexpr size_t OFF_T2  = OFF_A + 4194304;

// =============== weight repack: (Cout,Cin,K) -> (Cout,Cin,Kp) zero-padded ====
__global__ void pack_weights(const float* __restrict__ w, float* __restrict__ wp,
                             int total, int K, int Kp) {
  int i = blockIdx.x * blockDim.x + threadIdx.x;
  if (i >= total) return;
  int k = i % Kp;
  int cc = i / Kp;   // co*Cin + ci
  wp[i] = (k < K) ? w[cc * K + k] : 0.f;
}

// =============== conv1: Cin=1, Cout=16, K=17, W=8192 (direct) ===============
__global__ void conv1_direct(const float* __restrict__ x, const float* __restrict__ w,
                             float* __restrict__ y) {
  __shared__ float xs[272];
  __shared__ float ws[272];
  const int b = blockIdx.x;
  const int w0 = blockIdx.y * 256;
  const int t = threadIdx.x;
  for (int i = t; i < 272; i += 256) {
    ws[i] = w[i];
    int pos = w0 - 8 + i;
    xs[i] = (pos >= 0 && pos < 8192) ? x[(size_t)b * 8192 + pos] : 0.f;
  }
  __syncthreads();
  float win[17];
  for (int k = 0; k < 17; ++k) win[k] = xs[t + k];
  for (int c = 0; c < 16; ++c) {
    float acc = 0.f;
    for (int k = 0; k < 17; ++k) acc = fmaf(ws[c * 17 + k], win[k], acc);
    y[(size_t)b * 16 * 8192 + (size_t)c * 8192 + (size_t)(w0 + t)] = acc;
  }
}

// =============== generic WMMA conv: Y[b,co,w] = sum_{ci,k} W[co,ci,k]*X[b,ci,w+k-pad]
// one wave per 16(out-ch) x 16(position) tile, f32 WMMA 16x16x4.
// Wp is the zero-padded packed weight table (Cout, CIN, KP).
template <int CIN, int K, int KP>
__global__ void conv_wmma(const float* __restrict__ X, const float* __restrict__ Wp,
                          float* __restrict__ Y, int W, int Cout) {
  constexpr int TW = 16 + KP;
  constexpr int PAD = (K - 1) / 2;
  __shared__ float xs[CIN * TW];
  const int lane = threadIdx.x;
  const int wtiles = W / 16;
  const int b = blockIdx.x / wtiles;
  const int w0 = (blockIdx.x % wtiles) * 16;
  const int co0 = blockIdx.y * 16;

  const float* xb = X + (size_t)b * CIN * W;
  for (int i = lane; i < CIN * TW; i += 32) {
    int ci = i / TW, p = i % TW;
    int pos = w0 - PAD + p;
    xs[i] = (pos >= 0 && pos < W) ? xb[(size_t)ci * W + pos] : 0.f;
  }
  __syncthreads();

  const int m = lane & 15;           // A row (out-ch); also B col (position)
  const int bk = (lane >> 4) << 1;   // this lane holds K-offsets {bk, bk+1} of the 4-chunk
  const float* wr = Wp + (size_t)(co0 + m) * CIN * KP;
  v8f acc = {0.f, 0.f, 0.f, 0.f, 0.f, 0.f, 0.f, 0.f};

  for (int ci = 0; ci < CIN; ++ci) {
    const float* wrc = wr + ci * KP + bk;
    const float* xc = xs + ci * TW + m + bk;
    for (int kb = 0; kb < KP; kb += 4) {
      v2f a, bb;
      a.x = wrc[kb];
      a.y = wrc[kb + 1];
      bb.x = xc[kb];
      bb.y = xc[kb + 1];
      acc = __builtin_amdgcn_wmma_f32_16x16x4_f32(false, a, false, bb,
                                                  (short)0, acc, false, false);
    }
  }

  float* yb = Y + (size_t)b * Cout * W + (size_t)(w0 + m);
  const int mbase = (lane < 16) ? 0 : 8;
  for (int i = 0; i < 8; ++i) {
    int co = co0 + mbase + i;
    yb[(size_t)co * W] = acc[i];
  }
}

// =============== generic WMMA GEMM: C[m,n] = scale * sum_r A[m,r]*B[n,r] ==========
__global__ void gemm_abt_wmma(const float* __restrict__ A, const float* __restrict__ B,
                              float* __restrict__ C, int N, int Kd, float scale) {
  const int lane = threadIdx.x;
  const int n0 = blockIdx.x * 16;
  const int m0 = blockIdx.y * 16;
  const int mn = lane & 15;
  const int bk = (lane >> 4) << 1;
  const float* ar = A + (size_t)(m0 + mn) * Kd + bk;
  const float* br = B + (size_t)(n0 + mn) * Kd + bk;
  v8f acc = {0.f, 0.f, 0.f, 0.f, 0.f, 0.f, 0.f, 0.f};
  for (int r = 0; r < Kd; r += 4) {
    v2f a, b;
    a.x = ar[r]; a.y = ar[r + 1];
    b.x = br[r]; b.y = br[r + 1];
    acc = __builtin_amdgcn_wmma_f32_16x16x4_f32(false, a, false, b,
                                                (short)0, acc, false, false);
  }
  const int mbase = (lane < 16) ? 0 : 8;
  for (int i = 0; i < 8; ++i)
    C[(size_t)(m0 + mbase + i) * N + (size_t)(n0 + mn)] = acc[i] * scale;
}

// =============== helpers: stats / BN / pools ==================================
__global__ void zero_f32(float* p, int n) {
  int i = blockIdx.x * blockDim.x + threadIdx.x;
  if (i < n) p[i] = 0.f;
}

__global__ void reduce_stats(const float* __restrict__ y, float* __restrict__ stats,
                             int C, int W, int B) {
  __shared__ float s1[256], s2[256];
  const int c = blockIdx.x;
  const long long total = (long long)B * W;
  float s = 0.f, ss = 0.f;
  for (long long idx = (long long)blockIdx.y * 256 + threadIdx.x; idx < total;
       idx += (long long)gridDim.y * 256) {
    long long b = idx / W;
    long long w = idx - b * W;
    float v = y[(b * C + c) * (long long)W + w];
    s += v; ss += v * v;
  }
  s1[threadIdx.x] = s; s2[threadIdx.x] = ss;
  __syncthreads();
  for (int o = 128; o > 0; o >>= 1) {
    if (threadIdx.x < o) {
      s1[threadIdx.x] += s1[threadIdx.x + o];
      s2[threadIdx.x] += s2[threadIdx.x + o];
    }
    __syncthreads();
  }
  if (threadIdx.x == 0) {
    atomicAdd(&stats[c], s1[0]);
    atomicAdd(&stats[C + c], s2[0]);
  }
}

__global__ void bn_relu_pool4(const float* __restrict__ y, const float* __restrict__ stats,
                              const float* __restrict__ g, const float* __restrict__ be,
                              float* __restrict__ h, int C, int W, int B) {
  const int Wo = W / 4;
  long long i = (long long)blockIdx.x * blockDim.x + threadIdx.x;
  long long total = (long long)B * C * Wo;
  if (i >= total) return;
  int wo = (int)(i % Wo);
  long long bc = i / Wo;
  int c = (int)(bc % C);
  long long b = bc / C;
  float n = (float)B * (float)W;
  float mean = stats[c] / n;
  float var = stats[C + c] / n - mean * mean;
  float sc = g[c] * rsqrtf(var + EPS_BN);
  float sh = be[c] - mean * sc;
  const float* yp = y + (b * C + c) * (long long)W + (long long)wo * 4;
  float v0 = fmaxf(fmaf(yp[0], sc, sh), 0.f);
  float v1 = fmaxf(fmaf(yp[1], sc, sh), 0.f);
  float v2 = fmaxf(fmaf(yp[2], sc, sh), 0.f);
  float v3 = fmaxf(fmaf(yp[3], sc, sh), 0.f);
  h[i] = fmaxf(fmaxf(v0, v1), fmaxf(v2, v3));
}

__global__ void bn_relu_inplace(float* __restrict__ y, const float* __restrict__ stats,
                                const float* __restrict__ g, const float* __restrict__ be,
                                int C, int W, int B) {
  long long i = (long long)blockIdx.x * blockDim.x + threadIdx.x;
  long long total = (long long)B * C * W;
  if (i >= total) return;
  long long bc = i / W;
  int c = (int)(bc % C);
  float n = (float)B * (float)W;
  float mean = stats[c] / n;
  float var = stats[C + c] / n - mean * mean;
  float sc = g[c] * rsqrtf(var + EPS_BN);
  float sh = be[c] - mean * sc;
  y[i] = fmaxf(fmaf(y[i], sc, sh), 0.f);
}

// adaptive avg pool 128 -> 5 with torch bin boundaries
__global__ void adaptive_pool5(const float* __restrict__ t, float* __restrict__ k) {
  int i = blockIdx.x * blockDim.x + threadIdx.x;   // over B*64*5
  if (i >= 256 * 64 * 5) return;
  int q = i % 5;
  int bc = i / 5;
  int s = (q * 128) / 5;
  int e = ((q + 1) * 128 + 4) / 5;
  const float* p = t + (long long)bc * 128;
  float acc = 0.f;
  for (int j = s; j < e; ++j) acc += p[j];
  k[i] = acc / (float)(e - s);
}

// center+l2norm each (b,c) row of h (len 128); emit the 4-value S-pack:
// v = [h126+h127, h127, h0, h0+h1] (normalized); sim = (-k_pack)·v_pack / W
__global__ void norm_h_pack(const float* __restrict__ h, float* __restrict__ v) {
  __shared__ float red[128];
  __shared__ float edge[4];
  const int row = blockIdx.x;       // b*64 + c
  const int t = threadIdx.x;
  const float* p = h + (long long)row * 128;
  float val = p[t];
  red[t] = val; __syncthreads();
  for (int o = 64; o > 0; o >>= 1) { if (t < o) red[t] += red[t + o]; __syncthreads(); }
  float mean = red[0] / 128.f;
  __syncthreads();
  float cv = val - mean;
  red[t] = cv * cv; __syncthreads();
  for (int o = 64; o > 0; o >>= 1) { if (t < o) red[t] += red[t + o]; __syncthreads(); }
  float inv = 1.f / fmaxf(sqrtf(red[0]), EPS_NORM);
  if (t == 0) edge[0] = cv;
  if (t == 1) edge[1] = cv;
  if (t == 126) edge[2] = cv;
  if (t == 127) edge[3] = cv;
  __syncthreads();
  if (t == 0) {
    float* vp = v + (long long)row * 4;
    vp[0] = (edge[2] + edge[3]) * inv;
    vp[1] = edge[3] * inv;
    vp[2] = edge[0] * inv;
    vp[3] = (edge[0] + edge[1]) * inv;
  }
}

// center+l2norm each (b,c) row of k (len 5); emit negated pack at t = {0,1,3,4}
__global__ void norm_k_pack(const float* __restrict__ k, float* __restrict__ u) {
  int i = blockIdx.x * blockDim.x + threadIdx.x;   // B*64
  if (i >= 256 * 64) return;
  const float* p = k + (long long)i * 5;
  float m = (p[0] + p[1] + p[2] + p[3] + p[4]) * 0.2f;
  float c[5]; float ss = 0.f;
  for (int j = 0; j < 5; ++j) { c[j] = p[j] - m; ss += c[j] * c[j]; }
  float inv = 1.f / fmaxf(sqrtf(ss), EPS_NORM);
  float* up = u + (long long)i * 4;
  up[0] = -c[0] * inv;
  up[1] = -c[1] * inv;
  up[2] = -c[3] * inv;
  up[3] = -c[4] * inv;
}

// l2-normalize flat k rows (len 320) for the contrastive branch
__global__ void rownorm320(const float* __restrict__ k, float* __restrict__ a) {
  __shared__ float red[128];
  const int row = blockIdx.x;
  const int t = threadIdx.x;
  const float* p = k + (long long)row * 320;
  float ss = 0.f;
  for (int j = t; j < 320; j += 128) { float x = p[j]; ss += x * x; }
  red[t] = ss; __syncthreads();
  for (int o = 64; o > 0; o >>= 1) { if (t < o) red[t] += red[t + o]; __syncthreads(); }
  float inv = 1.f / fmaxf(sqrtf(red[0]), EPS_NORM);
  float* ap = a + (long long)row * 320;
  for (int j = t; j < 320; j += 128) ap[j] = p[j] * inv;
}

// per-row InfoNCE term: rl[i] = logsumexp(sim[i,:]/T) - sim[i,i]/T
__global__ void infonce_rows(const float* __restrict__ sim, float* __restrict__ rl) {
  __shared__ float red[256];
  const int i = blockIdx.x, t = threadIdx.x;
  float x = sim[(size_t)i * 256 + t] * INV_T;
  red[t] = x; __syncthreads();
  for (int o = 128; o > 0; o >>= 1) { if (t < o) red[t] = fmaxf(red[t], red[t + o]); __syncthreads(); }
  float mx = red[0]; __syncthreads();
  red[t] = __expf(x - mx); __syncthreads();
  for (int o = 128; o > 0; o >>= 1) { if (t < o) red[t] += red[t + o]; __syncthreads(); }
  if (t == 0) {
    float lse = mx + logf(red[0]);
    rl[i] = lse - sim[(size_t)i * 256 + i] * INV_T;
  }
}

__global__ void finalize_loss(const float* __restrict__ rl, float* __restrict__ out) {
  __shared__ float red[256];
  const int t = threadIdx.x;
  float sums[3];
  for (int j = 0; j < 3; ++j) {
    red[t] = rl[j * 256 + t];
    __syncthreads();
    for (int o = 128; o > 0; o >>= 1) { if (t < o) red[t] += red[t + o]; __syncthreads(); }
    sums[j] = red[0];
    __syncthreads();
  }
  if (t == 0)
    out[0] = 0.5f * (sums[0] + sums[1]) / 256.f + 0.5f * (sums[2] / 256.f);
}

// ===================== host-side pipeline ====================================
static void process_input(const float* x,
                          const float* w1, const float* g1, const float* be1,
                          const float* w2p, const float* g2, const float* be2,
                          const float* w3p, const float* g3, const float* be3,
                          const float* tw1p, const float* tg1, const float* tbe1,
                          const float* tw2p, const float* tg2, const float* tbe2,
                          float* ws, int which, hipStream_t stream) {
  float* stats = ws + OFF_ST;
  float* Y = ws + OFF_A;       // conv scratch
  float* H = ws + OFF_B;       // pooled scratch
  float* h3 = ws + OFF_H3 + (size_t)which * 2097152;
  float* kk = ws + OFF_K + (size_t)which * 81920;
  float* t1 = ws + OFF_T1;
  float* t2 = ws + OFF_T2;

  // layer 1: conv(1->16,K17) + BN + relu + pool4 : 8192 -> 2048
  conv1_direct<<<dim3(256, 32), 256, 0, stream>>>(x, w1, Y);
  zero_f32<<<1, 128, 0, stream>>>(stats, 128);
  reduce_stats<<<dim3(16, 64), 256, 0, stream>>>(Y, stats, 16, 8192, 256);
  bn_relu_pool4<<<32768, 256, 0, stream>>>(Y, stats, g1, be1, H, 16, 8192, 256);

  // layer 2: conv(16->32,K17) WMMA : W=2048 -> pool 512
  conv_wmma<16, 17, 20><<<dim3(256 * (2048 / 16), 2), 32, 0, stream>>>(H, w2p, Y, 2048, 32);
  zero_f32<<<1, 128, 0, stream>>>(stats, 128);
  reduce_stats<<<dim3(32, 64), 256, 0, stream>>>(Y, stats, 32, 2048, 256);
  bn_relu_pool4<<<16384, 256, 0, stream>>>(Y, stats, g2, be2, H, 32, 2048, 256);

  // layer 3: conv(32->64,K17) WMMA : W=512 -> pool 128
  conv_wmma<32, 17, 20><<<dim3(256 * (512 / 16), 4), 32, 0, stream>>>(H, w3p, Y, 512, 64);
  zero_f32<<<1, 128, 0, stream>>>(stats, 128);
  reduce_stats<<<dim3(64, 64), 256, 0, stream>>>(Y, stats, 64, 512, 256);
  bn_relu_pool4<<<8192, 256, 0, stream>>>(Y, stats, g3, be3, h3, 64, 512, 256);

  // template head: conv(64->64,K5) + BN + relu
  conv_wmma<64, 5, 8><<<dim3(256 * (128 / 16), 4), 32, 0, stream>>>(h3, tw1p, t1, 128, 64);
  zero_f32<<<1, 128, 0, stream>>>(stats, 128);
  reduce_stats<<<dim3(64, 64), 256, 0, stream>>>(t1, stats, 64, 128, 256);
  bn_relu_inplace<<<8192, 256, 0, stream>>>(t1, stats, tg1, tbe1, 64, 128, 256);

  // template head: conv(64->64,K3) + BN + relu
  conv_wmma<64, 3, 4><<<dim3(256 * (128 / 16), 4), 32, 0, stream>>>(t1, tw2p, t2, 128, 64);
  zero_f32<<<1, 128, 0, stream>>>(stats, 128);
  reduce_stats<<<dim3(64, 64), 256, 0, stream>>>(t2, stats, 64, 128, 256);
  bn_relu_inplace<<<8192, 256, 0, stream>>>(t2, stats, tg2, tbe2, 64, 128, 256);

  // adaptive pool -> k (B,64,5)
  adaptive_pool5<<<(256 * 64 * 5 + 255) / 256, 256, 0, stream>>>(t2, kk);

  // packs for the WMMA similarity / contrastive GEMMs
  norm_h_pack<<<16384, 128, 0, stream>>>(h3, ws + OFF_V + (size_t)which * 65536);
  norm_k_pack<<<64, 256, 0, stream>>>(kk, ws + OFF_U + (size_t)which * 65536);
  rownorm320<<<256, 128, 0, stream>>>(kk, ws + OFF_AN + (size_t)which * 81920);
}

extern "C" void kernel_launch(void* const* d_in, const int* in_sizes, int n_in,
                              void* d_out, int out_size, void* d_ws, size_t ws_size,
                              hipStream_t stream) {
  const float* x1 = (const float*)d_in[0];
  const float* x2 = (const float*)d_in[1];
  const float* w1 = (const float*)d_in[2];
  const float* g1 = (const float*)d_in[4];
  const float* be1 = (const float*)d_in[5];
  const float* w2 = (const float*)d_in[6];
  const float* g2 = (const float*)d_in[8];
  const float* be2 = (const float*)d_in[9];
  const float* w3 = (const float*)d_in[10];
  const float* g3 = (const float*)d_in[12];
  const float* be3 = (const float*)d_in[13];
  const float* tw1 = (const float*)d_in[14];
  const float* tg1 = (const float*)d_in[16];
  const float* tbe1 = (const float*)d_in[17];
  const float* tw2 = (const float*)d_in[18];
  const float* tg2 = (const float*)d_in[20];
  const float* tbe2 = (const float*)d_in[21];
  float* ws = (float*)d_ws;
  float* out = (float*)d_out;

  // one-time (per launch) zero-padded weight repack for guard-free WMMA loads
  float* w2p  = ws + OFF_W2P;
  float* w3p  = ws + OFF_W3P;
  float* tw1p = ws + OFF_TW1P;
  float* tw2p = ws + OFF_TW2P;
  pack_weights<<<(10240 + 255) / 256, 256, 0, stream>>>(w2, w2p, 10240, 17, 20);
  pack_weights<<<(40960 + 255) / 256, 256, 0, stream>>>(w3, w3p, 40960, 17, 20);
  pack_weights<<<(32768 + 255) / 256, 256, 0, stream>>>(tw1, tw1p, 32768, 5, 8);
  pack_weights<<<(16384 + 255) / 256, 256, 0, stream>>>(tw2, tw2p, 16384, 3, 4);

  process_input(x1, w1, g1, be1, w2p, g2, be2, w3p, g3, be3,
                tw1p, tg1, tbe1, tw2p, tg2, tbe2, ws, 0, stream);
  process_input(x2, w1, g1, be1, w2p, g2, be2, w3p, g3, be3,
                tw1p, tg1, tbe1, tw2p, tg2, tbe2, ws, 1, stream);

  float* u0 = ws + OFF_U, *u1 = ws + OFF_U + 65536;
  float* v0 = ws + OFF_V, *v1 = ws + OFF_V + 65536;
  float* a0 = ws + OFF_AN, *a1 = ws + OFF_AN + 81920;
  float* sim1 = ws + OFF_SIM, *sim2 = ws + OFF_SIM + 65536, *sim3 = ws + OFF_SIM + 131072;
  float* rl = ws + OFF_RL;

  // sim(k1,h2), sim(k2,h1): 256x256, K=256, scale 1/W (W=128)
  gemm_abt_wmma<<<dim3(16, 16), 32, 0, stream>>>(u0, v1, sim1, 256, 256, 1.f / 128.f);
  gemm_abt_wmma<<<dim3(16, 16), 32, 0, stream>>>(u1, v0, sim2, 256, 256, 1.f / 128.f);
  // contrastive: a·bT, K=320
  gemm_abt_wmma<<<dim3(16, 16), 32, 0, stream>>>(a0, a1, sim3, 256, 320, 1.f);

  infonce_rows<<<256, 256, 0, stream>>>(sim1, rl);
  infonce_rows<<<256, 256, 0, stream>>>(sim2, rl + 256);
  infonce_rows<<<256, 256, 0, stream>>>(sim3, rl + 512);
  finalize_loss<<<1, 256, 0, stream>>>(rl, out);
}